// Loss_20873541059058
// MI455X (gfx1250) — compile-verified
//
#include <hip/hip_runtime.h>
#include <math.h>

// Problem constants (from reference)
#define Bq 2048
#define Nq 4096
#define Dq 1024
#define CHq (Nq - 3)      // negatives chunk size per logits row
#define INV_TEMP 2.0f     // 1 / TEMP_F

typedef float v2f __attribute__((ext_vector_type(2)));
typedef float v8f __attribute__((ext_vector_type(8)));

// ---------------------------------------------------------------------------
// GEMM: sim = (H H^T) * INV_TEMP, H = [h_i ; h_j], fp32 via V_WMMA_F32_16X16X4_F32
// ---------------------------------------------------------------------------
#define BM 128
#define BN 128
#define BKq 32
#define LDT 36   // padded LDS row stride in floats (16B-aligned rows, conflict-free)

__device__ __forceinline__ const float* h_row(const float* hi, const float* hj, int r) {
  return (r < Bq) ? (hi + (size_t)r * Dq) : (hj + (size_t)(r - Bq) * Dq);
}

__global__ __launch_bounds__(256) void gemm_sim_kernel(const float* __restrict__ hi,
                                                       const float* __restrict__ hj,
                                                       float* __restrict__ sim) {
  __shared__ float As[BM * LDT];
  __shared__ float Bs[BN * LDT];
  const int rowBase = blockIdx.y * BM;
  const int colBase = blockIdx.x * BN;
  const int tid  = threadIdx.x;
  const int wave = tid >> 5;
  const int lane = tid & 31;
  const int wm = wave & 3;    // 0..3 -> 32-row strip
  const int wn = wave >> 2;   // 0..1 -> 64-col strip
  const int Ml = lane & 15;   // M (A) / N (B) index within 16
  const int kh = lane >> 4;   // K-half: lanes16-31 hold K=2,3

  v8f acc[2][4];
  v8f zero = {0.f,0.f,0.f,0.f,0.f,0.f,0.f,0.f};
  for (int rt = 0; rt < 2; ++rt)
    for (int ct = 0; ct < 4; ++ct) acc[rt][ct] = zero;

  for (int k0 = 0; k0 < Dq; k0 += BKq) {
    // Cooperative tile loads: 128x32 floats per tile = 1024 float4, 4 per thread
#pragma unroll
    for (int l = 0; l < 4; ++l) {
      int f  = tid + 256 * l;
      int rr = f >> 3;
      int kc = (f & 7) << 2;
      float4 av = *(const float4*)(h_row(hi, hj, rowBase + rr) + k0 + kc);
      *(float4*)&As[rr * LDT + kc] = av;
      float4 bv = *(const float4*)(h_row(hi, hj, colBase + rr) + k0 + kc);
      *(float4*)&Bs[rr * LDT + kc] = bv;
    }
    __syncthreads();

#pragma unroll
    for (int kk = 0; kk < BKq; kk += 4) {
      // A fragments (ISA 7.12.2: a[v] = A[M][2*kh+v])
      v2f a0 = *(const v2f*)&As[(wm * 32 +      Ml) * LDT + kk + 2 * kh];
      v2f a1 = *(const v2f*)&As[(wm * 32 + 16 + Ml) * LDT + kk + 2 * kh];
#pragma unroll
      for (int ct = 0; ct < 4; ++ct) {
        // B fragment: b[v] = B[2*kh+v][n] = H[col][k], same per-lane shape
        v2f b = *(const v2f*)&Bs[(wn * 64 + ct * 16 + Ml) * LDT + kk + 2 * kh];
        acc[0][ct] = __builtin_amdgcn_wmma_f32_16x16x4_f32(
            false, a0, false, b, (short)0, acc[0][ct], false, false);
        acc[1][ct] = __builtin_amdgcn_wmma_f32_16x16x4_f32(
            false, a1, false, b, (short)0, acc[1][ct], false, false);
      }
    }
    __syncthreads();
  }

  // C/D layout: VGPR g -> M = g + 8*kh, N = lane&15
#pragma unroll
  for (int rt = 0; rt < 2; ++rt)
#pragma unroll
    for (int ct = 0; ct < 4; ++ct) {
      int r0 = rowBase + wm * 32 + rt * 16 + kh * 8;
      int c  = colBase + wn * 64 + ct * 16 + Ml;
#pragma unroll
      for (int g = 0; g < 8; ++g)
        sim[(size_t)(r0 + g) * Nq + c] = acc[rt][ct][g] * INV_TEMP;
    }
}

// ---------------------------------------------------------------------------
// Top-4 mining per pair i (jax.lax.top_k semantics: ties -> lower index)
// ---------------------------------------------------------------------------
__device__ __forceinline__ void ins4(float* v, int* id, float s, int j) {
  if (s > v[3] || (s == v[3] && j < id[3])) {
    int p = 3;
    while (p > 0 && (s > v[p - 1] || (s == v[p - 1] && j < id[p - 1]))) {
      v[p] = v[p - 1]; id[p] = id[p - 1]; --p;
    }
    v[p] = s; id[p] = j;
  }
}

__global__ __launch_bounds__(256) void topk_kernel(const float* __restrict__ sim,
                                                   int* __restrict__ pos_cols,
                                                   int* __restrict__ pos_cnt) {
  const int i = blockIdx.x;
  const int t = threadIdx.x;
  float v[4]  = {-INFINITY, -INFINITY, -INFINITY, -INFINITY};
  int   id[4] = {0x7FFFFFFF, 0x7FFFFFFF, 0x7FFFFFFF, 0x7FFFFFFF};
  for (int j = t; j < 2 * Bq; j += 256) {
    float s = (j < Bq) ? sim[(size_t)i * Nq + Bq + j]
                       : sim[(size_t)(i + Bq) * Nq + (j - Bq)];
    ins4(v, id, s, j);
  }
  __shared__ float sv[256 * 4];
  __shared__ int   si[256 * 4];
  for (int x = 0; x < 4; ++x) { sv[t * 4 + x] = v[x]; si[t * 4 + x] = id[x]; }
  __syncthreads();
  if (t == 0) {
    float mv[4] = {-INFINITY, -INFINITY, -INFINITY, -INFINITY};
    int   mi[4] = {0x7FFFFFFF, 0x7FFFFFFF, 0x7FFFFFFF, 0x7FFFFFFF};
    for (int x = 0; x < 256 * 4; ++x) ins4(mv, mi, sv[x], si[x]);
    // select first TOPK=2 valid; map to sim cells
    int colsA[3], colsB[3]; int cA = 0, cB = 0;
    colsA[cA++] = i + Bq;   // cross positive of row i
    colsB[cB++] = i;        // cross positive of row i+B
    int taken = 0;
    for (int x = 0; x < 4 && taken < 2; ++x) {
      int j = mi[x];
      if (j == i || j == i + Bq) continue;
      ++taken;
      if (j < Bq) colsA[cA++] = j + Bq;   // row i,   col j+B
      else        colsB[cB++] = j - Bq;   // row i+B, col j-B
    }
    for (int a = 0; a < cA; ++a)
      for (int b2 = a + 1; b2 < cA; ++b2)
        if (colsA[b2] < colsA[a]) { int tm = colsA[a]; colsA[a] = colsA[b2]; colsA[b2] = tm; }
    for (int a = 0; a < cB; ++a)
      for (int b2 = a + 1; b2 < cB; ++b2)
        if (colsB[b2] < colsB[a]) { int tm = colsB[a]; colsB[a] = colsB[b2]; colsB[b2] = tm; }
    pos_cnt[i]      = cA;
    pos_cnt[i + Bq] = cB;
    for (int x = 0; x < 3; ++x) {
      pos_cols[3 * i + x]        = (x < cA) ? colsA[x] : 0x7FFFFFFF;
      pos_cols[3 * (i + Bq) + x] = (x < cB) ? colsB[x] : 0x7FFFFFFF;
    }
  }
}

// ---------------------------------------------------------------------------
// Exclusive prefix scan of per-row positive counts (single 256-thread block)
// ---------------------------------------------------------------------------
__global__ __launch_bounds__(256) void scan_kernel(const int* __restrict__ cnt,
                                                   int* __restrict__ excl) {
  __shared__ int part[256];
  const int t = threadIdx.x;
  const int base = t * 16;
  int local[16]; int s = 0;
  for (int x = 0; x < 16; ++x) { local[x] = s; s += cnt[base + x]; }
  part[t] = s; __syncthreads();
  for (int off = 1; off < 256; off <<= 1) {
    int vv = (t >= off) ? part[t - off] : 0;
    __syncthreads();
    part[t] += vv;
    __syncthreads();
  }
  int prev = (t == 0) ? 0 : part[t - 1];
  for (int x = 0; x < 16; ++x) excl[base + x] = prev + local[x];
}

// ---------------------------------------------------------------------------
// Per-logits-row max / sum-exp.  One block per sim row; a sim row touches at
// most 2 positive logits rows and 2 negative chunks -> 4 atomics per block.
// ---------------------------------------------------------------------------
__device__ __forceinline__ void atomicMaxF(unsigned int* addr, float val) {
  if (val >= 0.0f) atomicMax((int*)addr, __float_as_int(val));
  else             atomicMin(addr, __float_as_uint(val));
}

__global__ __launch_bounds__(256) void row_max_kernel(
    const float* __restrict__ sim, const int* __restrict__ pos_cols,
    const int* __restrict__ pos_cnt, const int* __restrict__ pos_excl,
    unsigned int* __restrict__ rowmax) {
  const int r = blockIdx.x, t = threadIdx.x;
  const int p = pos_cnt[r];
  const int pc0 = pos_cols[3 * r], pc1 = pos_cols[3 * r + 1], pc2 = pos_cols[3 * r + 2];
  const int qs  = pos_excl[r];
  const int kp0 = qs >> 1;
  const int negstart = r * (Nq - 1) - qs;
  const int kn0 = negstart / CHq;
  const int bnd = (kn0 + 1) * CHq;
  float mp0 = -INFINITY, mp1 = -INFINITY, mn0 = -INFINITY, mn1 = -INFINITY;
  for (int c = t; c < Nq; c += 256) {
    if (c == r) continue;
    float val = sim[(size_t)r * Nq + c];
    int x = (c == pc0) ? 0 : (c == pc1) ? 1 : (c == pc2) ? 2 : -1;
    if (x >= 0) {
      int q = qs + x;
      if ((q >> 1) == kp0) mp0 = fmaxf(mp0, val); else mp1 = fmaxf(mp1, val);
    } else {
      int less = (c > r ? 1 : 0) + (pc0 < c) + (pc1 < c) + (pc2 < c);
      int g = negstart + c - less;
      if (g < bnd) mn0 = fmaxf(mn0, val); else mn1 = fmaxf(mn1, val);
    }
  }
  __shared__ float sh[256];
  float loc[4] = {mp0, mp1, mn0, mn1}, red[4];
  for (int s2 = 0; s2 < 4; ++s2) {
    sh[t] = loc[s2]; __syncthreads();
    for (int off = 128; off > 0; off >>= 1) {
      if (t < off) sh[t] = fmaxf(sh[t], sh[t + off]);
      __syncthreads();
    }
    red[s2] = sh[0]; __syncthreads();
  }
  if (t == 0) {
    const int kp1 = (qs + p - 1) >> 1;
    atomicMaxF(&rowmax[kp0], red[0]);
    if (red[1] > -INFINITY) atomicMaxF(&rowmax[kp1], red[1]);
    atomicMaxF(&rowmax[kn0], red[2]);
    if (red[3] > -INFINITY && kn0 + 1 < Nq) atomicMaxF(&rowmax[kn0 + 1], red[3]);
  }
}

__global__ __launch_bounds__(256) void row_sum_kernel(
    const float* __restrict__ sim, const int* __restrict__ pos_cols,
    const int* __restrict__ pos_cnt, const int* __restrict__ pos_excl,
    const unsigned int* __restrict__ rowmax, float* __restrict__ rowsum,
    float* __restrict__ evenAcc) {
  const int r = blockIdx.x, t = threadIdx.x;
  const int p = pos_cnt[r];
  const int pc0 = pos_cols[3 * r], pc1 = pos_cols[3 * r + 1], pc2 = pos_cols[3 * r + 2];
  const int qs  = pos_excl[r];
  const int kp0 = qs >> 1;
  const int kp1 = (qs + p - 1) >> 1;
  const int negstart = r * (Nq - 1) - qs;
  const int kn0 = negstart / CHq;
  const int bnd = (kn0 + 1) * CHq;
  const float Mp0 = __uint_as_float(rowmax[kp0]);
  const float Mp1 = __uint_as_float(rowmax[kp1]);
  const float Mn0 = __uint_as_float(rowmax[kn0]);
  const float Mn1 = (kn0 + 1 < Nq) ? __uint_as_float(rowmax[kn0 + 1]) : 0.0f;
  float sp0 = 0.f, sp1 = 0.f, sn0 = 0.f, sn1 = 0.f, ev = 0.f;
  for (int c = t; c < Nq; c += 256) {
    if (c == r) continue;
    float val = sim[(size_t)r * Nq + c];
    int x = (c == pc0) ? 0 : (c == pc1) ? 1 : (c == pc2) ? 2 : -1;
    if (x >= 0) {
      int q = qs + x;
      if ((q >> 1) == kp0) sp0 += expf(val - Mp0); else sp1 += expf(val - Mp1);
      if (!(q & 1)) ev += val;   // logits[:,0] term
    } else {
      int less = (c > r ? 1 : 0) + (pc0 < c) + (pc1 < c) + (pc2 < c);
      int g = negstart + c - less;
      if (g < bnd) sn0 += expf(val - Mn0); else sn1 += expf(val - Mn1);
    }
  }
  __shared__ float sh[256];
  float loc[5] = {sp0, sp1, sn0, sn1, ev}, red[5];
  for (int s2 = 0; s2 < 5; ++s2) {
    sh[t] = loc[s2]; __syncthreads();
    for (int off = 128; off > 0; off >>= 1) {
      if (t < off) sh[t] += sh[t + off];
      __syncthreads();
    }
    red[s2] = sh[0]; __syncthreads();
  }
  if (t == 0) {
    atomicAdd(&rowsum[kp0], red[0]);
    if (kp1 != kp0) atomicAdd(&rowsum[kp1], red[1]);
    atomicAdd(&rowsum[kn0], red[2]);
    if (kn0 + 1 < Nq) atomicAdd(&rowsum[kn0 + 1], red[3]);
    atomicAdd(evenAcc, red[4]);
  }
}

__global__ void init_kernel(unsigned int* rowmax, float* rowsum, float* evenAcc) {
  int k = blockIdx.x * 256 + threadIdx.x;
  if (k < Nq) { rowmax[k] = 0xFF800000u; rowsum[k] = 0.0f; }
  if (k == 0) evenAcc[0] = 0.0f;
}

__global__ __launch_bounds__(256) void final_kernel(const unsigned int* __restrict__ rowmax,
                                                    const float* __restrict__ rowsum,
                                                    const float* __restrict__ evenAcc,
                                                    float* __restrict__ out) {
  const int t = threadIdx.x;
  float s = 0.f;
  for (int k = t; k < Nq; k += 256)
    s += __uint_as_float(rowmax[k]) + logf(rowsum[k]);
  __shared__ float sh[256];
  sh[t] = s; __syncthreads();
  for (int off = 128; off > 0; off >>= 1) {
    if (t < off) sh[t] += sh[t + off];
    __syncthreads();
  }
  if (t == 0) out[0] = (sh[0] - evenAcc[0]) / (float)Nq;
}

// ---------------------------------------------------------------------------
extern "C" void kernel_launch(void* const* d_in, const int* in_sizes, int n_in,
                              void* d_out, int out_size, void* d_ws, size_t ws_size,
                              hipStream_t stream) {
  const float* h_i = (const float*)d_in[0];
  const float* h_j = (const float*)d_in[1];
  float* out = (float*)d_out;

  char* ws = (char*)d_ws;
  size_t off = 0;
  float* sim = (float*)(ws + off);          off += (size_t)Nq * Nq * sizeof(float);
  int* pos_cols = (int*)(ws + off);         off += (size_t)Nq * 3 * sizeof(int);
  int* pos_cnt = (int*)(ws + off);          off += (size_t)Nq * sizeof(int);
  int* pos_excl = (int*)(ws + off);         off += (size_t)Nq * sizeof(int);
  unsigned int* rowmax = (unsigned int*)(ws + off); off += (size_t)Nq * sizeof(unsigned int);
  float* rowsum = (float*)(ws + off);       off += (size_t)Nq * sizeof(float);
  float* evenAcc = (float*)(ws + off);      off += 2 * sizeof(float);

  dim3 gemmGrid(Nq / BN, Nq / BM);
  gemm_sim_kernel<<<gemmGrid, 256, 0, stream>>>(h_i, h_j, sim);
  init_kernel<<<(Nq + 255) / 256, 256, 0, stream>>>(rowmax, rowsum, evenAcc);
  topk_kernel<<<Bq, 256, 0, stream>>>(sim, pos_cols, pos_cnt);
  scan_kernel<<<1, 256, 0, stream>>>(pos_cnt, pos_excl);
  row_max_kernel<<<Nq, 256, 0, stream>>>(sim, pos_cols, pos_cnt, pos_excl, rowmax);
  row_sum_kernel<<<Nq, 256, 0, stream>>>(sim, pos_cols, pos_cnt, pos_excl, rowmax,
                                         rowsum, evenAcc);
  final_kernel<<<1, 256, 0, stream>>>(rowmax, rowsum, evenAcc, out);
}